// CycleConsistencyLoss_26173530701950
// MI455X (gfx1250) — compile-verified
//
#include <hip/hip_runtime.h>
#include <hip/hip_bf16.h>

// Problem constants (from reference)
#define BB 4
#define NV 4
#define HW 36864          // 192*192
#define DD 64
#define AA 128
#define TT 4              // C(4,3)
#define THD 0.3f
#define MINVALID 5.0f
#define SEG 16
#define NTILES (HW / 16)              // 2304
#define TILES_PER_SEG (NTILES / SEG)  // 144 (even -> unroll x2)

typedef float v2f __attribute__((ext_vector_type(2)));
typedef float v8f __attribute__((ext_vector_type(8)));

__device__ __forceinline__ void get_triplet(int t, int& vi, int& vj, int& vk) {
    // combos of {0,1,2,3} choose 3, lexicographic:
    // t=0:(0,1,2) t=1:(0,1,3) t=2:(0,2,3) t=3:(1,2,3)
    vi = (t >= 3) ? 1 : 0;
    vj = (t >= 2) ? 2 : 1;
    vk = (t >= 1) ? 3 : 2;
}

__device__ __forceinline__ unsigned long long shflxor_u64(unsigned long long x, int m) {
    unsigned lo = (unsigned)x, hiw = (unsigned)(x >> 32);
    lo  = __shfl_xor(lo,  m, 32);
    hiw = __shfl_xor(hiw, m, 32);
    return ((unsigned long long)hiw << 32) | lo;
}

__global__ void init_kernel(unsigned long long* pij, unsigned long long* pjk) {
    int i = blockIdx.x * blockDim.x + threadIdx.x;
    if (i < BB * TT * AA) { pij[i] = ~0ull; pjk[i] = ~0ull; }
}

// Nearest-neighbor sweep: 128 query points vs one image's HW points.
// mode 0: queries = anchor points of view i, database = view j  (fills pack_ij)
// mode 1: queries = pj[argmin_ij] (from prev_pack), database = view k (fills pack_jk)
// One v_wmma_f32_16x16x4_f32 produces a full 16x16 tile of squared distances:
//   A row m = (Qx,Qy,Qz,1), B col n = (-2Px,-2Py,-2Pz,|P|^2), C[m][n] = |Q[m]|^2
//   -> D[m][n] = |Q|^2 + |P|^2 - 2 Q.P = d^2
// C is loop-invariant (built once via shuffles) so the WMMA needs no per-tile
// accumulator broadcast; only B changes per tile.
__global__ __launch_bounds__(256) void nn_kernel(
    const float* __restrict__ pts,              // (B,N,HW,3)
    const int* __restrict__ anchor_idx,         // (B,T,A)
    const unsigned long long* __restrict__ prev_pack,
    unsigned long long* __restrict__ out_pack,  // (B*T, A) packed (d2bits<<32)|idx
    int mode)
{
    const int bt  = blockIdx.x;          // 0..15
    const int seg = blockIdx.y;          // 0..SEG-1
    const int b = bt / TT, t = bt % TT;
    int vi, vj, vk; get_triplet(t, vi, vj, vk);
    const int qv = (mode == 0) ? vi : vj;
    const int dv = (mode == 0) ? vj : vk;

    const int wave = threadIdx.x >> 5;   // 0..7 -> query tile
    const int lane = threadIdx.x & 31;
    const int col  = lane & 15;
    const bool hi  = lane >= 16;

    // Query for this lane (A-matrix row = col within tile `wave`)
    const int q = wave * 16 + col;
    int qpix;
    if (mode == 0) qpix = anchor_idx[(b * TT + t) * AA + q];
    else           qpix = (int)(prev_pack[(size_t)bt * AA + q] & 0xffffffffull);

    const float* qp = pts + (((size_t)(b * NV + qv)) * HW + (size_t)qpix) * 3;
    const float qx = qp[0], qy = qp[1], qz = qp[2];
    const float nq = qx * qx + qy * qy + qz * qz;
    v2f amat;                     // lanes 0-15: K=0,1 ; lanes 16-31: K=2,3
    amat.x = hi ? qz : qx;
    amat.y = hi ? 1.0f : qy;

    // Loop-invariant C: c8[v] = |Q[row]|^2, row = wave*16 + (hi? 8+v : v).
    // nq for that row is held by the lane whose col == (hi? 8+v : v).
    v8f c8;
#pragma unroll
    for (int v = 0; v < 8; ++v) {
        const int src = (hi ? 24 : 0) + v;   // hi: lane 24+v has col 8+v
        c8[v] = __shfl(nq, src, 32);
    }

    const float* dbase = pts + ((size_t)(b * NV + dv)) * HW * 3;

    float bestD[8];
    int   bestN[8];
#pragma unroll
    for (int v = 0; v < 8; ++v) { bestD[v] = 3.402823e38f; bestN[v] = 0; }

    const int tile0 = seg * TILES_PER_SEG;
    for (int it = 0; it < TILES_PER_SEG; it += 2) {
        const int base0 = (tile0 + it) * 16;
        const int base1 = base0 + 16;
        // Issue both independent loads up front to overlap latency.
        const float* pp0 = dbase + (size_t)(base0 + col) * 3;
        const float* pp1 = dbase + (size_t)(base1 + col) * 3;
        const float p0x = pp0[0], p0y = pp0[1], p0z = pp0[2];
        const float p1x = pp1[0], p1y = pp1[1], p1z = pp1[2];

        {
            const float np = p0x * p0x + p0y * p0y + p0z * p0z;
            v2f bmat;
            bmat.x = hi ? (-2.0f * p0z) : (-2.0f * p0x);
            bmat.y = hi ? np            : (-2.0f * p0y);
            v8f d = __builtin_amdgcn_wmma_f32_16x16x4_f32(
                false, amat, false, bmat, (short)0, c8, false, false);
#pragma unroll
            for (int v = 0; v < 8; ++v) {
                if (d[v] < bestD[v]) { bestD[v] = d[v]; bestN[v] = base0 + col; }
            }
        }
        {
            const float np = p1x * p1x + p1y * p1y + p1z * p1z;
            v2f bmat;
            bmat.x = hi ? (-2.0f * p1z) : (-2.0f * p1x);
            bmat.y = hi ? np            : (-2.0f * p1y);
            v8f d = __builtin_amdgcn_wmma_f32_16x16x4_f32(
                false, amat, false, bmat, (short)0, c8, false, false);
#pragma unroll
            for (int v = 0; v < 8; ++v) {
                if (d[v] < bestD[v]) { bestD[v] = d[v]; bestN[v] = base1 + col; }
            }
        }
    }

    // Reduce min/argmin across the 16 lanes of each half (columns of the tile),
    // tie-break = lowest index via packed u64 lexicographic min.
#pragma unroll
    for (int v = 0; v < 8; ++v) {
        const float dc = fmaxf(bestD[v], 0.0f);  // keep f32 bit pattern monotone
        unsigned long long pk =
            ((unsigned long long)__float_as_uint(dc) << 32) | (unsigned)bestN[v];
#pragma unroll
        for (int m = 1; m < 16; m <<= 1) {
            unsigned long long o = shflxor_u64(pk, m);
            pk = (o < pk) ? o : pk;
        }
        if (col == 0) {  // lanes 0 (rows 0..7) and 16 (rows 8..15)
            const int qrow = wave * 16 + (hi ? 8 + v : v);
            atomicMin(out_pack + (size_t)bt * AA + qrow, pk);
        }
    }
}

// Per-(b,t): dki (128x128) via the same WMMA distance trick out of LDS,
// then the normalized-feature squared error and the masked mean.
__global__ __launch_bounds__(256) void finalize_kernel(
    const float* __restrict__ pts, const float* __restrict__ feats,
    const int* __restrict__ anchor_idx,
    const unsigned long long* __restrict__ pack_ij,
    const unsigned long long* __restrict__ pack_jk,
    float* __restrict__ loss_bt, float* __restrict__ cnt_bt)
{
    __shared__ float s_ap[AA][4];   // anchor points (x,y,z,|.|^2)
    __shared__ float s_kp[AA][4];   // chosen k-points
    __shared__ float s_mij[AA];
    __shared__ float s_mjk[AA];
    __shared__ float s_mki[AA];
    __shared__ int   s_ret[AA];
    __shared__ float s_w[8][2];

    const int bt = blockIdx.x;
    const int b = bt / TT, t = bt % TT;
    int vi, vj, vk; get_triplet(t, vi, vj, vk);
    const int tid = threadIdx.x;

    if (tid < AA) {   // waves 0..3 fully active: no intra-wave divergence
        const int a = tid;
        const int apix = anchor_idx[(b * TT + t) * AA + a];
        const float* ap = pts + (((size_t)(b * NV + vi)) * HW + (size_t)apix) * 3;
        const float x = ap[0], y = ap[1], z = ap[2];
        s_ap[a][0] = x; s_ap[a][1] = y; s_ap[a][2] = z; s_ap[a][3] = x*x + y*y + z*z;
        const unsigned long long pij = pack_ij[(size_t)bt * AA + a];
        s_mij[a] = sqrtf(__uint_as_float((unsigned)(pij >> 32)));
        const unsigned long long pjk = pack_jk[(size_t)bt * AA + a];
        s_mjk[a] = sqrtf(__uint_as_float((unsigned)(pjk >> 32)));
        const int kpix = (int)(pjk & 0xffffffffull);
        const float* kp = pts + (((size_t)(b * NV + vk)) * HW + (size_t)kpix) * 3;
        const float kx = kp[0], ky = kp[1], kz = kp[2];
        s_kp[a][0] = kx; s_kp[a][1] = ky; s_kp[a][2] = kz; s_kp[a][3] = kx*kx + ky*ky + kz*kz;
    }
    __syncthreads();

    const int wave = tid >> 5, lane = tid & 31, col = lane & 15;
    const bool hi = lane >= 16;
    const int q = wave * 16 + col;      // query = chosen k-point per anchor
    v2f amat;
    amat.x = hi ? s_kp[q][2] : s_kp[q][0];
    amat.y = hi ? 1.0f       : s_kp[q][1];

    // Loop-invariant C: query norms per D-matrix row, read straight from LDS.
    v8f c8;
#pragma unroll
    for (int v = 0; v < 8; ++v) {
        const int row = wave * 16 + (hi ? 8 + v : v);
        c8[v] = s_kp[row][3];
    }

    float bestD[8]; int bestN[8];
#pragma unroll
    for (int v = 0; v < 8; ++v) { bestD[v] = 3.402823e38f; bestN[v] = 0; }

#pragma unroll
    for (int tile = 0; tile < AA / 16; ++tile) {
        const int n = tile * 16 + col;
        v2f bmat;
        bmat.x = hi ? (-2.0f * s_ap[n][2]) : (-2.0f * s_ap[n][0]);
        bmat.y = hi ? s_ap[n][3]           : (-2.0f * s_ap[n][1]);
        v8f d = __builtin_amdgcn_wmma_f32_16x16x4_f32(
            false, amat, false, bmat, (short)0, c8, false, false);
#pragma unroll
        for (int v = 0; v < 8; ++v) {
            if (d[v] < bestD[v]) { bestD[v] = d[v]; bestN[v] = n; }
        }
    }
#pragma unroll
    for (int v = 0; v < 8; ++v) {
        const float dc = fmaxf(bestD[v], 0.0f);
        unsigned long long pk =
            ((unsigned long long)__float_as_uint(dc) << 32) | (unsigned)bestN[v];
#pragma unroll
        for (int m = 1; m < 16; m <<= 1) {
            unsigned long long o = shflxor_u64(pk, m);
            pk = (o < pk) ? o : pk;
        }
        if (col == 0) {
            const int qrow = wave * 16 + (hi ? 8 + v : v);
            s_mki[qrow] = sqrtf(__uint_as_float((unsigned)(pk >> 32)));
            s_ret[qrow] = (int)(pk & 0xffffffffull);
        }
    }
    __syncthreads();

    // se = na/Na^2 + nr/Nr^2 - 2*dot/(Na*Nr), Na = max(|a|,1e-12)
    float accSE = 0.0f, accC = 0.0f;
    for (int a = wave; a < AA; a += 8) {          // `a` uniform per wave
        const bool valid = (s_mij[a] < THD) & (s_mjk[a] < THD) & (s_mki[a] < THD);
        const int apix = anchor_idx[(b * TT + t) * AA + a];
        const int rpix = s_ret[a];                // raw pixel index in [0,A), per reference
        const float* fa = feats + (((size_t)(b * NV + vi)) * HW + (size_t)apix) * DD;
        const float* fr = feats + (((size_t)(b * NV + vi)) * HW + (size_t)rpix) * DD;
        const float a0 = fa[2 * lane], a1 = fa[2 * lane + 1];
        const float r0 = fr[2 * lane], r1 = fr[2 * lane + 1];
        float dot = a0 * r0 + a1 * r1;
        float na  = a0 * a0 + a1 * a1;
        float nr  = r0 * r0 + r1 * r1;
#pragma unroll
        for (int m = 1; m < 32; m <<= 1) {
            dot += __shfl_xor(dot, m, 32);
            na  += __shfl_xor(na,  m, 32);
            nr  += __shfl_xor(nr,  m, 32);
        }
        const float Na = fmaxf(sqrtf(na), 1e-12f);
        const float Nr = fmaxf(sqrtf(nr), 1e-12f);
        const float se = na / (Na * Na) + nr / (Nr * Nr) - 2.0f * dot / (Na * Nr);
        if (valid) { accSE += se; accC += 1.0f; }
    }
    if (lane == 0) { s_w[wave][0] = accSE; s_w[wave][1] = accC; }
    __syncthreads();
    if (tid == 0) {
        float sse = 0.0f, cnt = 0.0f;
        for (int w = 0; w < 8; ++w) { sse += s_w[w][0]; cnt += s_w[w][1]; }
        loss_bt[bt] = (cnt > 0.0f) ? sse / (fmaxf(cnt, 1.0f) * (float)DD) : 0.0f;
        cnt_bt[bt]  = cnt;
    }
}

__global__ void final_reduce(const float* __restrict__ loss_bt,
                             const float* __restrict__ cnt_bt,
                             float* __restrict__ out)
{
    if (threadIdx.x == 0 && blockIdx.x == 0) {
        float total = 0.0f;
        for (int t = 0; t < TT; ++t) {
            float tc = 0.0f, ls = 0.0f;
            for (int b = 0; b < BB; ++b) {
                const float c = cnt_bt[b * TT + t];
                if (c >= MINVALID) { tc += 1.0f; ls += loss_bt[b * TT + t]; }
            }
            total += (tc > 0.0f) ? ls / fmaxf(tc, 1.0f) : 0.0f;
        }
        out[0] = total / (float)TT;
    }
}

extern "C" void kernel_launch(void* const* d_in, const int* in_sizes, int n_in,
                              void* d_out, int out_size, void* d_ws, size_t ws_size,
                              hipStream_t stream) {
    (void)in_sizes; (void)n_in; (void)out_size; (void)ws_size;
    const float* feats = (const float*)d_in[0];   // (B,N,H,W,D) f32
    const float* pts   = (const float*)d_in[1];   // (B,N,H,W,3) f32
    const int*   aidx  = (const int*)d_in[2];     // (B,T,A) i32

    unsigned long long* pack_ij = (unsigned long long*)d_ws;
    unsigned long long* pack_jk = pack_ij + (size_t)BB * TT * AA;
    float* loss_bt = (float*)(pack_jk + (size_t)BB * TT * AA);
    float* cnt_bt  = loss_bt + BB * TT;

    init_kernel<<<(BB * TT * AA + 255) / 256, 256, 0, stream>>>(pack_ij, pack_jk);

    dim3 grid(BB * TT, SEG);
    nn_kernel<<<grid, 256, 0, stream>>>(pts, aidx, pack_ij, pack_ij, 0);
    nn_kernel<<<grid, 256, 0, stream>>>(pts, aidx, pack_ij, pack_jk, 1);

    finalize_kernel<<<BB * TT, 256, 0, stream>>>(pts, feats, aidx, pack_ij, pack_jk,
                                                 loss_bt, cnt_bt);
    final_reduce<<<1, 32, 0, stream>>>(loss_bt, cnt_bt, (float*)d_out);
}